// CoxSGDLossFn_44951127720573
// MI455X (gfx1250) — compile-verified
//
#include <hip/hip_runtime.h>
#include <hip/hip_bf16.h>

#define NROWS 8192
#define SENT  (-1e30f)   // sentinel value for ineligible / empty slots
#define SENTC (-1e29f)   // strictly above SENT, strictly below any rand >= 0

typedef float v4f __attribute__((ext_vector_type(4)));

// Insert candidate (v, j) into descending top-3 value set (v1>=v2>=v3),
// indices tracked for top-2 only. Value updates use v_max/v_med3 (exact).
__device__ __forceinline__ void topk_ins(float v, int j,
                                         float& v1, int& j1,
                                         float& v2, int& j2,
                                         float& v3) {
  const bool g1 = v > v1;
  const bool g2 = v > v2;
  const float nv1 = fmaxf(v, v1);
  const float nv2 = __builtin_amdgcn_fmed3f(v, v1, v2);
  const float nv3 = __builtin_amdgcn_fmed3f(v, v2, v3);
  j2 = g1 ? j1 : (g2 ? j : j2);
  j1 = g1 ? j  : j1;
  v1 = nv1; v2 = nv2; v3 = nv3;
}

// Kernel A: per-row masked top-3/top-2 scan of rand_mat.
// One wave (32 lanes) per row; 8 rows per 256-thread block.
// length[] staged into LDS via CDNA5 async global->LDS copies (ASYNCcnt).
__global__ __launch_bounds__(256) void cox_topk_scan(
    const float* __restrict__ length,
    const float* __restrict__ event,
    const float* __restrict__ rand_mat,
    int* __restrict__ rec) {
  __shared__ float s_len[NROWS];  // 32 KB
  const unsigned t = threadIdx.x;

  // ---- CDNA5 async copy: length (8192 f32 = 32KB) global -> LDS ----
  {
    const unsigned base = (unsigned)(size_t)(&s_len[0]);
#pragma unroll
    for (int k = 0; k < 8; ++k) {
      const unsigned e4  = (unsigned)k * 256u + t;       // float4 index
      const unsigned lds = base + e4 * 16u;
      const unsigned long long g =
          (unsigned long long)(size_t)(length + (size_t)e4 * 4u);
      asm volatile("global_load_async_to_lds_b128 %0, %1, off"
                   :: "v"(lds), "v"(g) : "memory");
    }
    asm volatile("s_wait_asynccnt 0x0" ::: "memory");
  }
  __syncthreads();

  const int lane = (int)(t & 31u);
  const int wave = (int)(t >> 5u);
  const int i = (int)blockIdx.x * 8 + wave;

  // event[i]==0 => entire pair row is zero: nothing contributes.
  if (event[i] == 0.0f) {
    if (lane == 0) rec[i] = 0;
    return;
  }

  const float leni = s_len[i];

  float v1 = SENT, v2 = SENT, v3 = SENT;
  int   j1 = 0, j2 = 0;

  const v4f* __restrict__ prow =
      (const v4f*)(rand_mat + (size_t)i * (size_t)NROWS);

#pragma unroll 4
  for (int it = 0; it < NROWS / 128; ++it) {
    const int col0 = it * 128 + lane * 4;
    // single-use stream: non-temporal 128-bit load
    const v4f r4 = __builtin_nontemporal_load(prow + (it * 32 + lane));
    const v4f l4 = *(const v4f*)(&s_len[col0]);
#pragma unroll
    for (int q = 0; q < 4; ++q) {
      // length[j] > length[i] ? rand : sentinel (sentinel never displaces)
      const float v = (l4[q] > leni) ? r4[q] : SENT;
      topk_ins(v, col0 + q, v1, j1, v2, j2, v3);
    }
  }

  // wave32 butterfly merge of per-lane top-3 sets.
  // Partner's 3rd value can never reach merged rank <= 2, so it only
  // competes for v3 (no index needed).
#pragma unroll
  for (int m = 16; m >= 1; m >>= 1) {
    const float bv1 = __shfl_xor(v1, m, 32);
    const int   bj1 = __shfl_xor(j1, m, 32);
    const float bv2 = __shfl_xor(v2, m, 32);
    const int   bj2 = __shfl_xor(j2, m, 32);
    const float bv3 = __shfl_xor(v3, m, 32);
    topk_ins(bv1, bj1, v1, j1, v2, j2, v3);
    topk_ins(bv2, bj2, v1, j1, v2, j2, v3);
    v3 = fmaxf(v3, bv3);
  }

  if (lane == 0) {
    // v_k > SENTC  <=>  at least k eligible entries in the row.
    // thr = 3rd-largest p (or below-all if <3 eligible); strict '>' keeps
    // exactly the reference's post-filter set, including tie cases.
    const float thr = (v3 > SENTC) ? v3 : SENTC;
    const int kept1 = (v1 > thr) ? 1 : 0;
    const int kept2 = (v2 > thr) ? 1 : 0;
    const int valid = kept1;  // post-filter row-sum > 0  <=>  top-1 kept
    rec[i] = valid | (kept1 << 1) | (kept2 << 2) | (j1 << 3) | (j2 << 16);
  }
}

// Kernel B: single-block deterministic finalize.
__global__ __launch_bounds__(1024) void cox_finalize(
    const float* __restrict__ y_pred,
    const int* __restrict__ rec,
    float* __restrict__ out) {
  __shared__ int   s_cnt[NROWS];   // 32 KB column counts
  __shared__ float s_red[1024];
  const int t = threadIdx.x;

  // ---- M = max(y_pred), fixed-order tree reduction ----
  float m = -3.4e38f;
#pragma unroll
  for (int k = 0; k < 8; ++k) m = fmaxf(m, y_pred[t + 1024 * k]);
  s_red[t] = m;
  __syncthreads();
  for (int s = 512; s > 0; s >>= 1) {
    if (t < s) s_red[t] = fmaxf(s_red[t], s_red[t + s]);
    __syncthreads();
  }
  const float M = s_red[0];
  __syncthreads();

#pragma unroll
  for (int k = 0; k < 8; ++k) s_cnt[t + 1024 * k] = 0;
  __syncthreads();

  // ---- per-row loss + integer column counts (exact -> deterministic) ----
  float lsum = 0.0f;
#pragma unroll
  for (int k = 0; k < 8; ++k) {
    const int i = t + 1024 * k;
    const int r = rec[i];
    if (r & 1) {
      const int jj1 = (r >> 3)  & 8191;
      const int jj2 = (r >> 16) & 8191;
      float rs = expf(y_pred[i] - M);        // diagonal (pair[i,i]=valid)
      atomicAdd(&s_cnt[i], 1);
      if (r & 2) { rs += expf(y_pred[jj1] - M); atomicAdd(&s_cnt[jj1], 1); }
      if (r & 4) { rs += expf(y_pred[jj2] - M); atomicAdd(&s_cnt[jj2], 1); }
      lsum += (M - y_pred[i]) + logf(rs);    // row_max + log(row_sum)
    }
  }
  __syncthreads();

  // ---- regularizer = sum_j |colcnt[j] * y_pred[j]| ----
  float reg = 0.0f;
#pragma unroll
  for (int k = 0; k < 8; ++k) {
    const int j = t + 1024 * k;
    reg += fabsf((float)s_cnt[j] * y_pred[j]);
  }

  // ---- fixed-order sum reductions ----
  s_red[t] = lsum;
  __syncthreads();
  for (int s = 512; s > 0; s >>= 1) {
    if (t < s) s_red[t] += s_red[t + s];
    __syncthreads();
  }
  const float L = s_red[0];
  __syncthreads();

  s_red[t] = reg;
  __syncthreads();
  for (int s = 512; s > 0; s >>= 1) {
    if (t < s) s_red[t] += s_red[t + s];
    __syncthreads();
  }
  if (t == 0) out[0] = L + 0.05f * s_red[0];
}

extern "C" void kernel_launch(void* const* d_in, const int* in_sizes, int n_in,
                              void* d_out, int out_size, void* d_ws, size_t ws_size,
                              hipStream_t stream) {
  (void)in_sizes; (void)n_in; (void)out_size; (void)ws_size;
  const float* y_pred   = (const float*)d_in[0];
  const float* length   = (const float*)d_in[1];
  const float* event    = (const float*)d_in[2];
  const float* rand_mat = (const float*)d_in[3];
  int* rec = (int*)d_ws;  // NROWS ints = 32 KB, fully rewritten each call

  cox_topk_scan<<<dim3(NROWS / 8), dim3(256), 0, stream>>>(
      length, event, rand_mat, rec);
  cox_finalize<<<dim3(1), dim3(1024), 0, stream>>>(
      y_pred, rec, (float*)d_out);
}